// NN_Custom_torch_2_89043261981085
// MI455X (gfx1250) — compile-verified
//
#include <hip/hip_runtime.h>
#include <hip/hip_bf16.h>

typedef __attribute__((ext_vector_type(2))) float v2f;
typedef __attribute__((ext_vector_type(8))) float v8f;

#define LW_STRIDE 68   // 64 + 4 pad: row r starts at bank 4r -> conflict-free b64 A reads

// GEMV via V_WMMA_F32_16X16X4_F32, double-buffered through LDS.
// Grid: (M/128, KSPLIT). Block: 256 threads (8 waves, 16 rows/wave).
// Pipeline per 128x64 f32 tile:
//   issue next tile's coalesced global_load_b128 into VGPRs
//   -> run 16 WMMAs on current LDS buffer (A = W rows, B = x broadcast to 16 cols)
//   -> commit registers to the other LDS buffer.
// Writes per-K-split partials: zpart[blockIdx.y * M + row].
__global__ __launch_bounds__(256)
void gemv_wmma_f32(const float* __restrict__ W, const float* __restrict__ x,
                   float* __restrict__ zpart, int M, int K, int kchunk) {
    __shared__ float lw[2][128 * LW_STRIDE];
    __shared__ float lx[2][64];

    const int t    = threadIdx.x;
    const int lane = t & 31;
    const int wave = t >> 5;
    const int row0 = blockIdx.x * 128;
    const int kbeg = blockIdx.y * kchunk;

    // --- hoisted addressing: one base pointer + fixed 16-row stride ---
    const int    rl  = t >> 4;           // 0..15
    const int    cl  = (t & 15) * 4;     // 0..60
    const float* wp  = W + (size_t)(row0 + rl) * K + kbeg + cl;
    const size_t rs16 = (size_t)16 * K;  // 16-row stride in floats
    const float* xp  = x + kbeg + (t & 15) * 4;

    const int rloc  = lane & 15;         // row within the wave's 16-row group
    const int khalf = lane >> 4;         // 0 -> K pair {0,1}; 1 -> K pair {2,3}
    const int arow_off = (wave * 16 + rloc) * LW_STRIDE;

    v8f acc = {};
    float4 r[8];
    float4 xv;

    // ---- prologue: fetch tile 0 and commit to buffer 0 ----
#pragma unroll
    for (int i = 0; i < 8; ++i) r[i] = *(const float4*)(wp + i * rs16);
    if (t < 16) xv = *(const float4*)(xp);
#pragma unroll
    for (int i = 0; i < 8; ++i) *(float4*)&lw[0][(rl + 16 * i) * LW_STRIDE + cl] = r[i];
    if (t < 16) *(float4*)&lx[0][cl] = xv;
    __syncthreads();

    const int ntiles = kchunk >> 6;      // kchunk / 64
    for (int it = 0; it < ntiles; ++it) {
        const int cur = it & 1;
        const int nxt = cur ^ 1;

        // ---- issue next tile's loads (in flight during WMMA phase) ----
        if (it + 1 < ntiles) {
            const float* wpn = wp + (size_t)(it + 1) * 64;
#pragma unroll
            for (int i = 0; i < 8; ++i) r[i] = *(const float4*)(wpn + i * rs16);
            if (t < 16) xv = *(const float4*)(xp + (it + 1) * 64);
        }

        // ---- 16 WMMAs cover the current K tile of 64 (K=4 each) ----
        const float* arow = &lw[cur][arow_off];
        const float* xc   = &lx[cur][0];
#pragma unroll
        for (int kk = 0; kk < 64; kk += 4) {
            const int kb = kk + khalf * 2;
            v2f a; a.x = arow[kb]; a.y = arow[kb + 1];   // A 16x4 lane layout
            v2f b; b.x = xc[kb];   b.y = xc[kb + 1];     // B 4x16, all N columns = x
            acc = __builtin_amdgcn_wmma_f32_16x16x4_f32(
                /*neg_a=*/false, a, /*neg_b=*/false, b,
                /*c_mod=*/(short)0, acc, /*reuse_a=*/false, /*reuse_b=*/false);
        }
        __syncthreads();

        // ---- commit prefetched registers to the other buffer ----
        if (it + 1 < ntiles) {
#pragma unroll
            for (int i = 0; i < 8; ++i) *(float4*)&lw[nxt][(rl + 16 * i) * LW_STRIDE + cl] = r[i];
            if (t < 16) *(float4*)&lx[nxt][cl] = xv;
            __syncthreads();
        }
    }

    // D layout: VGPR r -> (lanes 0..15: M=r, N=lane) (lanes 16..31: M=8+r, N=lane-16).
    // Every column holds the same dot product; take N=0 from lanes 0 and 16.
    if ((lane & 15) == 0) {
        float* outp = zpart + (size_t)blockIdx.y * M + row0 + wave * 16 + khalf * 8;
#pragma unroll
        for (int rr = 0; rr < 8; ++rr) outp[rr] = acc[rr];
    }
}

// out[m] (or out[m] +=) tanh( LN( sum_s zpart[s*M + m] ) * g + b ), biased variance.
// Single block of 256 threads; M in {1024, 4096} -> 4..16 elements per thread.
__global__ __launch_bounds__(256)
void ln_tanh_f32(const float* __restrict__ zpart, int nsplit, int M,
                 const float* __restrict__ g, const float* __restrict__ bb,
                 float* __restrict__ out, int accumulate) {
    __shared__ float s1[256], s2[256], stats[2];
    const int t = threadIdx.x;
    const int per = M >> 8;              // M / 256 (<= 16)

    float zv[16];
    float lsum = 0.f, lsq = 0.f;
    for (int i = 0; i < per; ++i) {
        const int m = t + (i << 8);
        float z = 0.f;
        for (int s = 0; s < nsplit; ++s) z += zpart[(size_t)s * M + m];
        zv[i] = z; lsum += z; lsq += z * z;
    }
    s1[t] = lsum; s2[t] = lsq;
    __syncthreads();
    for (int w = 128; w > 0; w >>= 1) {
        if (t < w) { s1[t] += s1[t + w]; s2[t] += s2[t + w]; }
        __syncthreads();
    }
    if (t == 0) {
        const float mean = s1[0] / (float)M;
        const float var  = s2[0] / (float)M - mean * mean;
        stats[0] = mean;
        stats[1] = rsqrtf(var + 1e-5f);
    }
    __syncthreads();
    const float mean = stats[0], inv = stats[1];
    for (int i = 0; i < per; ++i) {
        const int m = t + (i << 8);
        const float v = tanhf((zv[i] - mean) * inv * g[m] + bb[m]);
        if (accumulate) out[m] += v; else out[m] = v;
    }
}

extern "C" void kernel_launch(void* const* d_in, const int* in_sizes, int n_in,
                              void* d_out, int out_size, void* d_ws, size_t ws_size,
                              hipStream_t stream) {
    const float* x_in  = (const float*)d_in[0];   // input_raw (2048)
    const float* h2p   = (const float*)d_in[1];   // h2_prev   (4096)
    const float* W_I_H1  = (const float*)d_in[2]; // 4096x2048
    const float* W_H2_H1 = (const float*)d_in[3]; // 4096x4096
    const float* W_H1_H2 = (const float*)d_in[4]; // 4096x4096
    const float* W_H1_O  = (const float*)d_in[5]; // 1024x4096
    const float* W_H2_O  = (const float*)d_in[6]; // 1024x4096
    const float* g1 = (const float*)d_in[7];
    const float* b1 = (const float*)d_in[8];
    const float* g2 = (const float*)d_in[9];
    const float* b2 = (const float*)d_in[10];
    const float* gO = (const float*)d_in[11];
    const float* bO = (const float*)d_in[12];
    float* out = (float*)d_out;                   // 1024 f32

    // workspace layout (floats)
    float* ws    = (float*)d_ws;
    float* zbufA = ws;                 // up to 8 splits x 4096
    float* zbufB = ws + 8 * 4096;      // up to 8 splits x 4096
    float* h1    = ws + 16 * 4096;     // 4096
    float* h2    = h1 + 4096;          // 4096
    float* oacc  = h2 + 4096;          // 1024
    (void)ws_size; (void)n_in; (void)in_sizes; (void)out_size;

    const dim3 blk(256);

    // z1a = W_I_H1 @ x   (M=4096, K=2048, 4 K-splits of 512)
    gemv_wmma_f32<<<dim3(4096 / 128, 4), blk, 0, stream>>>(W_I_H1, x_in, zbufA, 4096, 2048, 512);
    // z1b = W_H2_H1 @ h2_prev  (M=4096, K=4096, 8 K-splits of 512)
    gemv_wmma_f32<<<dim3(4096 / 128, 8), blk, 0, stream>>>(W_H2_H1, h2p, zbufB, 4096, 4096, 512);
    // h1 = tanh(LN(z1a)) + tanh(LN(z1b))
    ln_tanh_f32<<<1, blk, 0, stream>>>(zbufA, 4, 4096, g1, b1, h1, 0);
    ln_tanh_f32<<<1, blk, 0, stream>>>(zbufB, 8, 4096, g1, b1, h1, 1);

    // z2 = W_H1_H2 @ h1 ; h2 = tanh(LN(z2))
    gemv_wmma_f32<<<dim3(4096 / 128, 8), blk, 0, stream>>>(W_H1_H2, h1, zbufA, 4096, 4096, 512);
    ln_tanh_f32<<<1, blk, 0, stream>>>(zbufA, 8, 4096, g2, b2, h2, 0);

    // za = W_H1_O @ h1 ; zb = W_H2_O @ h2
    gemv_wmma_f32<<<dim3(1024 / 128, 8), blk, 0, stream>>>(W_H1_O, h1, zbufA, 1024, 4096, 512);
    gemv_wmma_f32<<<dim3(1024 / 128, 8), blk, 0, stream>>>(W_H2_O, h2, zbufB, 1024, 4096, 512);

    // o_acc = tanh(LN(za)) + tanh(LN(zb)) ; out = tanh(LN(o_acc))
    ln_tanh_f32<<<1, blk, 0, stream>>>(zbufA, 8, 1024, gO, bO, oacc, 0);
    ln_tanh_f32<<<1, blk, 0, stream>>>(zbufB, 8, 1024, gO, bO, oacc, 1);
    ln_tanh_f32<<<1, blk, 0, stream>>>(oacc, 1, 1024, gO, bO, out, 0);
}